// Part_Block_67156108640586
// MI455X (gfx1250) — compile-verified
//
#include <hip/hip_runtime.h>
#include <math.h>

typedef __attribute__((ext_vector_type(2))) float v2f;
typedef __attribute__((ext_vector_type(8))) float v8f;

#define B_  32
#define C_  2048
#define HW_ 576        // 24*24
#define P_  4
#define HID_ 128       // C/RED

// ---------------------------------------------------------------------------
// Stage 1: pooled[b,c] = mean over HW of x[b,c,:,:]   (one wave per (b,c))
// ---------------------------------------------------------------------------
__global__ __launch_bounds__(256) void pool_kernel(const float* __restrict__ x,
                                                   float* __restrict__ pooled) {
  int gw   = (blockIdx.x * 256 + threadIdx.x) >> 5;   // 0 .. B*C-1
  int lane = threadIdx.x & 31;
  const float* src = x + (size_t)gw * HW_;
  float s = 0.f;
  #pragma unroll
  for (int i = lane; i < HW_; i += 32) s += src[i];
  #pragma unroll
  for (int off = 16; off > 0; off >>= 1) s += __shfl_xor(s, off, 32);
  if (lane == 0) pooled[gw] = s * (1.0f / (float)HW_);
}

// ---------------------------------------------------------------------------
// Stage 2a: hidden[b,j] = silu(pooled[b,:] . fc1_w[j,:] + fc1_b[j])
//           one wave per (b,j); 32*128 = 4096 waves
// ---------------------------------------------------------------------------
__global__ __launch_bounds__(256) void fc1_kernel(const float* __restrict__ pooled,
                                                  const float* __restrict__ fc1_w,
                                                  const float* __restrict__ fc1_b,
                                                  float* __restrict__ hidden) {
  int gw   = (blockIdx.x * 256 + threadIdx.x) >> 5;   // 0 .. 4095
  int lane = threadIdx.x & 31;
  int b = gw >> 7, j = gw & 127;
  const float* pv = pooled + (size_t)b * C_;
  const float* wv = fc1_w + (size_t)j * C_;
  float s = 0.f;
  for (int k = lane; k < C_; k += 32) s = fmaf(pv[k], wv[k], s);
  #pragma unroll
  for (int off = 16; off > 0; off >>= 1) s += __shfl_xor(s, off, 32);
  if (lane == 0) {
    float h = s + fc1_b[j];
    hidden[b * HID_ + j] = h / (1.0f + expf(-h));     // silu
  }
}

// ---------------------------------------------------------------------------
// Stage 2b: dw[b,p,c] = hidden[b,:] . fc2_w[p*C+c,:] + fc2_b[p*C+c]
//           wtil_pad[b, m, c] (m = 0..15):  m<4 -> dw*conv_w[c], else 0
//           One block of 256 threads handles 256 (m,c) outputs for one b.
// ---------------------------------------------------------------------------
__global__ __launch_bounds__(256) void fc2_kernel(const float* __restrict__ hidden,
                                                  const float* __restrict__ fc2_w,
                                                  const float* __restrict__ fc2_b,
                                                  const float* __restrict__ conv_w,
                                                  float* __restrict__ dw,
                                                  float* __restrict__ wtil_pad) {
  __shared__ float hsh[HID_];
  int bid   = blockIdx.x;           // 0 .. 32*128-1
  int b     = bid >> 7;             // 0 .. 31
  int chunk = bid & 127;            // 0 .. 127
  if (threadIdx.x < HID_) hsh[threadIdx.x] = hidden[b * HID_ + threadIdx.x];
  __syncthreads();
  int i = chunk * 256 + threadIdx.x;     // 0 .. 16*2048-1  (padded rows)
  int p = i >> 11;                       // 0 .. 15
  int c = i & (C_ - 1);
  if (p < P_) {
    int row = p * C_ + c;
    const float* wr = fc2_w + (size_t)row * HID_;
    float s = fc2_b[row];
    #pragma unroll 8
    for (int k = 0; k < HID_; ++k) s = fmaf(hsh[k], wr[k], s);
    dw[((size_t)(b * P_ + p)) * C_ + c]        = s;
    wtil_pad[((size_t)(b * 16 + p)) * C_ + c]  = s * conv_w[c];
  } else {
    wtil_pad[((size_t)(b * 16 + p)) * C_ + c]  = 0.f;   // zero-pad rows 4..15
  }
}

// ---------------------------------------------------------------------------
// Stage 2c: bias_term[b,p] = conv_b . dw[b,p,:]     (one wave per (b,p))
// ---------------------------------------------------------------------------
__global__ __launch_bounds__(256) void bias_kernel(const float* __restrict__ dw,
                                                   const float* __restrict__ conv_b,
                                                   float* __restrict__ bias_term) {
  int gw   = (blockIdx.x * 256 + threadIdx.x) >> 5;   // 0 .. 127
  int lane = threadIdx.x & 31;
  const float* dv = dw + (size_t)gw * C_;
  float s = 0.f;
  for (int c = lane; c < C_; c += 32) s = fmaf(conv_b[c], dv[c], s);
  #pragma unroll
  for (int off = 16; off > 0; off >>= 1) s += __shfl_xor(s, off, 32);
  if (lane == 0) bias_term[gw] = s;
}

// ---------------------------------------------------------------------------
// Stage 3: out[b,p,hw] = softmax_p( sum_c x[b,c,hw]*wtil[b,p,c] + bias[b,p] )
// Per wave: one 16x16 D tile via V_WMMA_F32_16X16X4_F32, K-loop over C.
//   A (16x4 f32):  lane l holds M=l&15; vgpr0/1 = K=koff,koff+1 (koff=(l>=16)?2:0)
//   B (4x16 f32):  lane l holds N=l&15; vgpr0/1 = K=koff,koff+1
//   D (16x16 f32): lanes 0-15 hold N=lane, vgprs 0..3 = rows M=0..3 (the 4 parts)
// 32 batches * 36 column tiles = 1152 waves.
// ---------------------------------------------------------------------------
__global__ __launch_bounds__(256) void einsum_kernel(const float* __restrict__ x,
                                                     const float* __restrict__ wtil_pad,
                                                     const float* __restrict__ bias_term,
                                                     float* __restrict__ out) {
  int gw   = (blockIdx.x * 256 + threadIdx.x) >> 5;   // 0 .. 1151
  int lane = threadIdx.x & 31;
  int b    = gw / 36;
  int tile = gw - b * 36;
  int n0   = tile * 16;

  int M    = lane & 15;
  int koff = (lane >> 4) << 1;                        // 0 or 2
  int col  = n0 + M;

  const float* xb   = x + (size_t)b * C_ * HW_;
  const float* arow = wtil_pad + ((size_t)(b * 16 + M)) * C_ + koff;  // rows 4..15 are zero

  v8f acc = {};
  #pragma unroll 4
  for (int k = 0; k < C_; k += 4) {
    v2f a;
    a.x = arow[k];
    a.y = arow[k + 1];
    v2f bv;
    bv.x = xb[(size_t)(k + koff) * HW_ + col];
    bv.y = xb[(size_t)(k + koff + 1) * HW_ + col];
    acc = __builtin_amdgcn_wmma_f32_16x16x4_f32(
        /*neg_a=*/false, a, /*neg_b=*/false, bv,
        /*c_mod=*/(short)0, acc, /*reuse_a=*/false, /*reuse_b=*/false);
  }

  if (lane < 16) {
    float p0 = acc[0] + bias_term[b * P_ + 0];
    float p1 = acc[1] + bias_term[b * P_ + 1];
    float p2 = acc[2] + bias_term[b * P_ + 2];
    float p3 = acc[3] + bias_term[b * P_ + 3];
    float m  = fmaxf(fmaxf(p0, p1), fmaxf(p2, p3));
    float e0 = expf(p0 - m), e1 = expf(p1 - m), e2 = expf(p2 - m), e3 = expf(p3 - m);
    float inv = 1.0f / (e0 + e1 + e2 + e3);
    out[((size_t)(b * P_ + 0)) * HW_ + col] = e0 * inv;
    out[((size_t)(b * P_ + 1)) * HW_ + col] = e1 * inv;
    out[((size_t)(b * P_ + 2)) * HW_ + col] = e2 * inv;
    out[((size_t)(b * P_ + 3)) * HW_ + col] = e3 * inv;
  }
}

// ---------------------------------------------------------------------------
extern "C" void kernel_launch(void* const* d_in, const int* in_sizes, int n_in,
                              void* d_out, int out_size, void* d_ws, size_t ws_size,
                              hipStream_t stream) {
  const float* x      = (const float*)d_in[0];
  const float* fc1_w  = (const float*)d_in[1];
  const float* fc1_b  = (const float*)d_in[2];
  const float* fc2_w  = (const float*)d_in[3];
  const float* fc2_b  = (const float*)d_in[4];
  const float* conv_w = (const float*)d_in[5];
  const float* conv_b = (const float*)d_in[6];
  float* out = (float*)d_out;

  // workspace layout (floats)
  float* ws        = (float*)d_ws;
  float* pooled    = ws;                               // 32*2048      = 65536
  float* hidden    = pooled + B_ * C_;                 // 32*128       = 4096
  float* dw        = hidden + B_ * HID_;               // 32*4*2048    = 262144
  float* wtil_pad  = dw + B_ * P_ * C_;                // 32*16*2048   = 1048576
  float* bias_term = wtil_pad + B_ * 16 * C_;          // 32*4         = 128

  // Stage 1: pooling  (65536 waves, 8 per block)
  pool_kernel<<<(B_ * C_) / 8, 256, 0, stream>>>(x, pooled);
  // Stage 2a: fc1 + silu (4096 waves)
  fc1_kernel<<<(B_ * HID_) / 8, 256, 0, stream>>>(pooled, fc1_w, fc1_b, hidden);
  // Stage 2b: fc2 -> dw, wtil_pad (zero-padded to 16 rows per batch)
  fc2_kernel<<<B_ * 128, 256, 0, stream>>>(hidden, fc2_w, fc2_b, conv_w, dw, wtil_pad);
  // Stage 2c: bias_term (128 waves)
  bias_kernel<<<(B_ * P_) / 8, 256, 0, stream>>>(dw, conv_b, bias_term);
  // Stage 3: WMMA contraction + softmax (1152 waves)
  einsum_kernel<<<(B_ * 36) / 8, 256, 0, stream>>>(x, wtil_pad, bias_term, out);
}